// LSTM_5712306503951
// MI455X (gfx1250) — compile-verified
//
#include <hip/hip_runtime.h>
#include <math.h>

typedef __attribute__((ext_vector_type(2))) float v2f;
typedef __attribute__((ext_vector_type(8))) float v8f;

#define B_DIM 256
#define S_DIM 4096
#define I_DIM 32
#define ROW_STRIDE 33               // I + 1 (value channel appended)
#define NROWS (B_DIM * S_DIM)       // 1,048,576 (b-major rows of x)
#define NTILES (NROWS / 16)         // 65,536 M-tiles of 16 rows

__device__ __forceinline__ float sigm_f(float x) {
  return 1.0f / (1.0f + __expf(-x));
}
__device__ __forceinline__ float tanh_f(float x) {
  // tanh(x) = 1 - 2/(1+e^{2x}); correct limits at +-inf with fast exp
  return 1.0f - 2.0f / (1.0f + __expf(2.0f * x));
}

// ---------------------------------------------------------------------------
// Phase 1: xgT[(s*B+b)*4+g] = sum_k x[b,s,k]*W_ih0[g,k]  (bias folded later)
//          vt[s*B+b]        = x[b,s,32]
// One wave per 16-row tile; K=32 via 8x V_WMMA_F32_16X16X4_F32.
// ---------------------------------------------------------------------------
__global__ __launch_bounds__(256) void lstm_proj_wmma(
    const float* __restrict__ x, const float* __restrict__ Wih0,
    float* __restrict__ xgT, float* __restrict__ vt)
{
  __shared__ float lds[8][16 * ROW_STRIDE];   // per-wave tile, pad 33 -> no bank conflicts

  const int lane = threadIdx.x & 31;
  const int wid  = threadIdx.x >> 5;
  const int n    = lane & 15;   // N (gate col) for B/D frags; M row for A frag
  const int hi   = lane >> 4;   // K half select per ISA f32 A/B layout

  // B fragments (K x N = 4x16, rows g>=4 zero-padded), one per K4 step.
  v2f bf[8];
#pragma unroll
  for (int s8 = 0; s8 < 8; ++s8) {
    float w0 = 0.0f, w1 = 0.0f;
    if (n < 4) {
      const int k0 = 4 * s8 + 2 * hi;
      w0 = Wih0[n * I_DIM + k0];
      w1 = Wih0[n * I_DIM + k0 + 1];
    }
    bf[s8].x = w0;
    bf[s8].y = w1;
  }

  const int nWaves = (gridDim.x * blockDim.x) >> 5;                 // 4096
  const int wave   = (blockIdx.x * blockDim.x + threadIdx.x) >> 5;
  float* tile = &lds[wid][0];

  for (int t = wave; t < NTILES; t += nWaves) {   // uniform 16 iters/wave
    const int m_base = t * 16;

    // Coalesced cooperative load: lane = column, iterate 16 rows.
#pragma unroll 4
    for (int r = 0; r < 16; ++r)
      tile[r * ROW_STRIDE + lane] =
          x[(size_t)(m_base + r) * ROW_STRIDE + lane];
    __asm__ volatile("s_wait_dscnt 0x0" ::: "memory");

    v8f c = {};
#pragma unroll
    for (int s8 = 0; s8 < 8; ++s8) {
      const int k0 = 4 * s8 + 2 * hi;
      v2f a;                                     // 16x4 f32 A frag
      a.x = tile[n * ROW_STRIDE + k0];
      a.y = tile[n * ROW_STRIDE + k0 + 1];
      c = __builtin_amdgcn_wmma_f32_16x16x4_f32(false, a, false, bf[s8],
                                                (short)0, c, false, false);
    }

    // D layout: VGPR j -> M=j (lanes 0-15) / M=j+8 (lanes 16-31), N=lane&15.
    if (n < 4) {
#pragma unroll
      for (int j = 0; j < 8; ++j) {
        const int rm = m_base + j + 8 * hi;      // global row (b-major)
        const int rb = rm >> 12;                 // / S
        const int rs = rm & (S_DIM - 1);         // % S
        xgT[((size_t)rs * B_DIM + rb) * 4 + n] = c[j];
      }
    }
    __asm__ volatile("" ::: "memory");           // keep LDS reuse ordered
  }

  // Value-channel transpose gather: coalesced writes to vt[S,B].
  const int tid = blockIdx.x * blockDim.x + threadIdx.x;
  const int nT  = gridDim.x * blockDim.x;
  for (int idx = tid; idx < NROWS; idx += nT) {
    const int b = idx & (B_DIM - 1);
    const int s = idx >> 8;
    vt[idx] = x[((size_t)b * S_DIM + s) * ROW_STRIDE + I_DIM];
  }
}

// ---------------------------------------------------------------------------
// Phase 2: fused 2-layer H=1 LSTM scan + online softmax-weighted sum + sigmoid
// One batch element per lane; 8 single-wave blocks (latency-bound serial scan).
// ---------------------------------------------------------------------------
__global__ __launch_bounds__(32) void lstm_scan(
    const float* __restrict__ xgT, const float* __restrict__ vt,
    const float* __restrict__ Whh0, const float* __restrict__ bih0,
    const float* __restrict__ bhh0, const float* __restrict__ Wih1,
    const float* __restrict__ Whh1, const float* __restrict__ bih1,
    const float* __restrict__ bhh1, float* __restrict__ out)
{
  const int b = blockIdx.x * 32 + threadIdx.x;

  float wh0[4], b0[4], wi1[4], wh1[4], b1[4];
#pragma unroll
  for (int g = 0; g < 4; ++g) {
    wh0[g] = Whh0[g];
    b0[g]  = bih0[g] + bhh0[g];
    wi1[g] = Wih1[g];
    wh1[g] = Whh1[g];
    b1[g]  = bih1[g] + bhh1[g];
  }

  float h0 = 0.0f, c0 = 0.0f, h1 = 0.0f, c1 = 0.0f;
  float m = -INFINITY, ssum = 0.0f, wsum = 0.0f;
  const float4* xg4 = (const float4*)xgT;

  for (int t = 0; t < S_DIM; ++t) {
    if (t + 16 < S_DIM) {
      __builtin_prefetch(&xg4[(size_t)(t + 16) * B_DIM + b], 0, 0);
      __builtin_prefetch(&vt[(size_t)(t + 16) * B_DIM + b], 0, 0);
    }
    const float4 xg = xg4[(size_t)t * B_DIM + b];   // coalesced 16B/lane
    const float val = vt[(size_t)t * B_DIM + b];    // coalesced 4B/lane

    // Layer 0 cell (gate order i,f,g,o)
    const float i0 = sigm_f(xg.x + h0 * wh0[0] + b0[0]);
    const float f0 = sigm_f(xg.y + h0 * wh0[1] + b0[1]);
    const float g0 = tanh_f(xg.z + h0 * wh0[2] + b0[2]);
    const float o0 = sigm_f(xg.w + h0 * wh0[3] + b0[3]);
    c0 = f0 * c0 + i0 * g0;
    h0 = o0 * tanh_f(c0);

    // Layer 1 cell
    const float i1 = sigm_f(h0 * wi1[0] + h1 * wh1[0] + b1[0]);
    const float f1 = sigm_f(h0 * wi1[1] + h1 * wh1[1] + b1[1]);
    const float g1 = tanh_f(h0 * wi1[2] + h1 * wh1[2] + b1[2]);
    const float o1 = sigm_f(h0 * wi1[3] + h1 * wh1[3] + b1[3]);
    c1 = f1 * c1 + i1 * g1;
    h1 = o1 * tanh_f(c1);

    // Online softmax over t, fused with weighted sum of the value channel.
    const float mn   = fmaxf(m, h1);
    const float corr = __expf(m - mn);
    const float e    = __expf(h1 - mn);
    ssum = ssum * corr + e;
    wsum = wsum * corr + e * val;
    m = mn;
  }

  out[b] = sigm_f(wsum / ssum);
}

extern "C" void kernel_launch(void* const* d_in, const int* in_sizes, int n_in,
                              void* d_out, int out_size, void* d_ws, size_t ws_size,
                              hipStream_t stream) {
  const float* x    = (const float*)d_in[0];
  const float* Wih0 = (const float*)d_in[1];
  const float* Whh0 = (const float*)d_in[2];
  const float* bih0 = (const float*)d_in[3];
  const float* bhh0 = (const float*)d_in[4];
  const float* Wih1 = (const float*)d_in[5];
  const float* Whh1 = (const float*)d_in[6];
  const float* bih1 = (const float*)d_in[7];
  const float* bhh1 = (const float*)d_in[8];
  float* out = (float*)d_out;

  float* xgT = (float*)d_ws;                          // S*B*4 f32 = 16 MB
  float* vt  = xgT + (size_t)S_DIM * B_DIM * 4;       // S*B   f32 =  4 MB

  lstm_proj_wmma<<<512, 256, 0, stream>>>(x, Wih0, xgT, vt);
  lstm_scan<<<8, 32, 0, stream>>>(xgT, vt, Whh0, bih0, bhh0,
                                  Wih1, Whh1, bih1, bhh1, out);
}